// CombinedLoss_81527069213369
// MI455X (gfx1250) — compile-verified
//
#include <hip/hip_runtime.h>
#include <cstdint>
#include <cstddef>

// Problem constants (fixed by the reference harness)
#define NIMG   16
#define HH     1024
#define WW     1024
#define IGN    255
#define TR     8              // rows per block tile
#define BLOCK  256            // 8 waves of 32
#define NWAVES 8
#define ROWBLKS (HH / TR)          // 128
#define NBLOCKS (NIMG * ROWBLKS)   // 2048

static_assert(TR * 32 == BLOCK, "phase B mapping requires TR*32 == BLOCK");

#if defined(__HIP_DEVICE_COMPILE__) && __has_builtin(__builtin_amdgcn_tensor_load_to_lds) && __has_builtin(__builtin_amdgcn_s_wait_tensorcnt)
#define USE_TDM 1
#else
#define USE_TDM 0
#endif

typedef unsigned int u32x4 __attribute__((ext_vector_type(4)));
typedef int          i32x4 __attribute__((ext_vector_type(4)));
typedef int          i32x8 __attribute__((ext_vector_type(8)));

__device__ __forceinline__ unsigned ballot32(bool p) {
#if __has_builtin(__builtin_amdgcn_ballot_w32)
  return __builtin_amdgcn_ballot_w32(p);
#else
  return (unsigned)__ballot(p);
#endif
}

__global__ __launch_bounds__(BLOCK) void loss_partial_kernel(
    const float* __restrict__ logits,   // [N,2,H,W]
    const int*   __restrict__ targets,  // [N,H,W]
    float* __restrict__ partials)       // [NBLOCKS,3] : focal, weighted-ce, count
{
  __shared__ unsigned roadW[(TR + 4) * 32];   // road bit-words, rows with +-2 halo
  __shared__ unsigned ignW[TR * 32];          // ignore bit-words, tile rows
  __shared__ unsigned vorW[TR * 32];          // vertical OR (dilation, 1D)
  __shared__ unsigned vandW[TR * 32];         // vertical AND (erosion, 1D)
  __shared__ unsigned bndW[TR * 32];          // boundary bit-words
  __shared__ float    red[NWAVES * 3];
#if USE_TDM
  extern __shared__ float ldsLog[];           // [2][TR][WW] = 64 KB, filled by TDM
#endif

  const int tid  = threadIdx.x;
  const int lane = tid & 31;
  const int wave = tid >> 5;
  const int blk  = blockIdx.x;
  const int n    = blk / ROWBLKS;
  const int row0 = (blk % ROWBLKS) * TR;

  const size_t imgBase = (size_t)n * 2 * HH * WW;  // image n, channel 0 element offset

#if USE_TDM
  // ---- Kick off TDM: stream the 2-channel logits tile into LDS (wave 0 only,
  //      so the DMA is issued once per block). Overlaps with phases A/B below.
  if (wave == 0) {
    const unsigned ldsBase = (unsigned)(size_t)(&ldsLog[0]);  // LDS flat addr low bits = offset
    for (int c = 0; c < 2; ++c) {
      unsigned long long gaddr =
          (unsigned long long)(const void*)(logits + imgBase + (size_t)c * HH * WW +
                                            (size_t)row0 * WW);
      u32x4 g0;
      g0[0] = 1u;                                               // count=1 (valid), user mode
      g0[1] = ldsBase + (unsigned)(c * TR * WW * 4);            // lds_addr (bytes)
      g0[2] = (unsigned)(gaddr & 0xffffffffu);                  // global_addr[31:0]
      g0[3] = (unsigned)((gaddr >> 32) & 0x01ffffffu) | (2u << 30); // addr[56:32] | type=2
      i32x8 g1;
      g1[0] = (2 << 16);                                        // data_size=4B, no pad/iter/mcast
      g1[1] = (int)((WW & 0xffff) << 16);                       // tensor_dim0 lo16 @bit48
      g1[2] = (int)(((WW >> 16) & 0xffff) | ((HH & 0xffff) << 16)); // dim0 hi | dim1 lo
      g1[3] = (int)(((HH >> 16) & 0xffff) | ((WW & 0xffff) << 16)); // dim1 hi | tile_dim0=WW
      g1[4] = TR;                                               // tile_dim1=TR, tile_dim2=0
      g1[5] = WW;                                               // tensor_dim0_stride lo32
      g1[6] = 0;
      g1[7] = 0;
      i32x4 gz = {0, 0, 0, 0};
#if defined(__clang_major__) && (__clang_major__ >= 23)
      i32x8 gz8 = {0, 0, 0, 0, 0, 0, 0, 0};
      __builtin_amdgcn_tensor_load_to_lds(g0, g1, gz, gz, gz8, 0);
#else
      __builtin_amdgcn_tensor_load_to_lds(g0, g1, gz, gz, 0);
#endif
    }
  }
#endif

  // ---- Phase A: pack road / ignore bits via wave ballot (coalesced B32 loads).
  for (int task = wave; task < (TR + 4) * 32; task += NWAVES) {
    const int rr = task >> 5;            // halo-relative row 0..TR+3
    const int w  = task & 31;            // 32-column word index
    const int y  = row0 - 2 + rr;
    unsigned roadBits = 0u, ignBits = 0u;
    if ((unsigned)y < (unsigned)HH) {
      const int t = targets[((size_t)n * HH + y) * WW + w * 32 + lane];
      roadBits = ballot32(t == 1);
      ignBits  = ballot32(t == IGN);
    }
    if (lane == 0) {
      roadW[task] = roadBits;
      if (rr >= 2 && rr < TR + 2) ignW[(rr - 2) * 32 + w] = ignBits;
    }
  }
  __syncthreads();

  // ---- Phase B1: vertical OR/AND over 5 rows (clip rows = SAME padding with
  //      -inf/+inf identities). One word per thread.
  {
    const int ry = tid >> 5, w = tid & 31;
    const int y  = row0 + ry;
    unsigned vo = 0u, va = 0xffffffffu;
#pragma unroll
    for (int dy = -2; dy <= 2; ++dy) {
      const int yy = y + dy;
      if ((unsigned)yy < (unsigned)HH) {
        const unsigned u = roadW[(ry + dy + 2) * 32 + w];
        vo |= u;
        va &= u;
      }
    }
    vorW[tid]  = vo;
    vandW[tid] = va;
  }
  __syncthreads();

  // ---- Phase B2: horizontal +-2 with word carries; edges shift in 0 (OR) / 1 (AND).
  {
    const int w = tid & 31;
    const unsigned c = vorW[tid];
    const unsigned l = (w > 0)  ? vorW[tid - 1] : 0u;
    const unsigned r = (w < 31) ? vorW[tid + 1] : 0u;
    const unsigned dil = c | (c << 1) | (l >> 31) | (c << 2) | (l >> 30)
                           | (c >> 1) | (r << 31) | (c >> 2) | (r << 30);
    const unsigned ca = vandW[tid];
    const unsigned la = (w > 0)  ? vandW[tid - 1] : 0xffffffffu;
    const unsigned ra = (w < 31) ? vandW[tid + 1] : 0xffffffffu;
    const unsigned ero = ca & ((ca << 1) | (la >> 31)) & ((ca << 2) | (la >> 30))
                            & ((ca >> 1) | (ra << 31)) & ((ca >> 2) | (ra << 30));
    bndW[tid] = dil & ~ero;
  }

#if USE_TDM
  if (wave == 0) __builtin_amdgcn_s_wait_tensorcnt(0);   // TDM tile landed in LDS
#endif
  __syncthreads();

  // ---- Phase C: per-pixel CE / focal / weighted-CE; float4 per thread per row.
  float accF = 0.f, accW = 0.f, accC = 0.f;
#if !USE_TDM
  const float* log0 = logits + imgBase + (size_t)row0 * WW;
  const float* log1 = log0 + (size_t)HH * WW;
#endif
  const int x0i = tid * 4;
  const int w   = x0i >> 5;
  const int sh  = x0i & 31;
#pragma unroll
  for (int ry = 0; ry < TR; ++ry) {
    const unsigned road = roadW[(ry + 2) * 32 + w] >> sh;
    const unsigned ign  = ignW[ry * 32 + w] >> sh;
    const unsigned bnd  = bndW[ry * 32 + w] >> sh;
#if USE_TDM
    const float4 l0 = ((const float4*)(ldsLog + (size_t)ry * WW))[tid];
    const float4 l1 = ((const float4*)(ldsLog + (size_t)(TR + ry) * WW))[tid];
#else
    const float4 l0 = ((const float4*)(log0 + (size_t)ry * WW))[tid];
    const float4 l1 = ((const float4*)(log1 + (size_t)ry * WW))[tid];
#endif
    const float a0[4] = {l0.x, l0.y, l0.z, l0.w};
    const float a1[4] = {l1.x, l1.y, l1.z, l1.w};
#pragma unroll
    for (int j = 0; j < 4; ++j) {
      const float x0 = a0[j], x1 = a1[j];
      const float mx = fmaxf(x0, x1);
      const float mn = fminf(x0, x1);
      const float xt = ((road >> j) & 1u) ? x1 : x0;
      const float ce = (mx - xt) + log1pf(expf(mn - mx));   // -log_softmax[t]
      const float pt = expf(-ce);
      const float om = 1.f - pt;
      const float fo = om * om * ce;                        // gamma = 2
      const float v  = ((ign >> j) & 1u) ? 0.f : 1.f;
      const float wt = 1.f + (float)((bnd >> j) & 1u);      // 1 + boundary*(B_EMPH-1)
      accF += fo * v;
      accW += ce * wt * v;
      accC += v;
    }
  }

  // ---- Block reduction (deterministic): wave shuffle tree, then wave partials.
#pragma unroll
  for (int off = 16; off > 0; off >>= 1) {
    accF += __shfl_xor(accF, off, 32);
    accW += __shfl_xor(accW, off, 32);
    accC += __shfl_xor(accC, off, 32);
  }
  if (lane == 0) {
    red[wave * 3 + 0] = accF;
    red[wave * 3 + 1] = accW;
    red[wave * 3 + 2] = accC;
  }
  __syncthreads();
  if (tid == 0) {
    float sF = 0.f, sW = 0.f, sC = 0.f;
    for (int i = 0; i < NWAVES; ++i) {
      sF += red[i * 3 + 0];
      sW += red[i * 3 + 1];
      sC += red[i * 3 + 2];
    }
    partials[(size_t)blk * 3 + 0] = sF;
    partials[(size_t)blk * 3 + 1] = sW;
    partials[(size_t)blk * 3 + 2] = sC;
  }
}

__global__ __launch_bounds__(256) void loss_final_kernel(
    const float* __restrict__ partials, float* __restrict__ out)
{
  __shared__ double s[256][3];
  double f = 0.0, wce = 0.0, c = 0.0;
  for (int i = threadIdx.x; i < NBLOCKS; i += 256) {
    f   += (double)partials[(size_t)i * 3 + 0];
    wce += (double)partials[(size_t)i * 3 + 1];
    c   += (double)partials[(size_t)i * 3 + 2];
  }
  s[threadIdx.x][0] = f;
  s[threadIdx.x][1] = wce;
  s[threadIdx.x][2] = c;
  __syncthreads();
  for (int off = 128; off > 0; off >>= 1) {
    if (threadIdx.x < off) {
      s[threadIdx.x][0] += s[threadIdx.x + off][0];
      s[threadIdx.x][1] += s[threadIdx.x + off][1];
      s[threadIdx.x][2] += s[threadIdx.x + off][2];
    }
    __syncthreads();
  }
  if (threadIdx.x == 0) {
    const double nv = s[0][2];
    out[0] = (float)((1.0 * s[0][0] + 0.5 * s[0][1]) / nv);  // FOCAL_W=1, BOUND_W=0.5
  }
}

extern "C" void kernel_launch(void* const* d_in, const int* in_sizes, int n_in,
                              void* d_out, int out_size, void* d_ws, size_t ws_size,
                              hipStream_t stream) {
  (void)in_sizes; (void)n_in; (void)out_size; (void)ws_size;
  const float* logits  = (const float*)d_in[0];
  const int*   targets = (const int*)d_in[1];
  float* partials = (float*)d_ws;                 // NBLOCKS*3 floats = 24 KB scratch
  // Always reserve the 64 KB dynamic-LDS logits tile (device decides whether the
  // TDM path consumes it; harmless if the fallback path is compiled).
  const size_t shmem = (size_t)2 * TR * WW * sizeof(float);
  loss_partial_kernel<<<NBLOCKS, BLOCK, shmem, stream>>>(logits, targets, partials);
  loss_final_kernel<<<1, 256, 0, stream>>>(partials, (float*)d_out);
}